// RadiusGraph_89781996355966
// MI455X (gfx1250) — compile-verified
//
#include <hip/hip_runtime.h>

// Radius graph with per-atom K-nearest cap, fused for MI455X (gfx1250).
//   - seg_kernel: binary-search molecule segment boundaries (batch is sorted)
//   - radius_kernel: one block per molecule; positions LDS-resident; distance
//     tiles via V_WMMA_F32_16X16X4_F32; top-K via wave32 shuffle reduction.
//
// Wave id is readfirstlane'd into an SGPR so every loop that encloses the
// WMMA is scalar-controlled: EXEC is structurally all-1s at the WMMA site
// (ISA 7.12 requirement), and the compiler drops the divergent-loop
// save/restore-exec bookkeeping seen in the previous build.

#define N_ATOMS  8192
#define N_MOLS   64
#define K_NEI    32
#define CUTOFF2  25.0f
#define MPAD     512                 // max padded molecule size (avg ~128)
#define NK       (N_ATOMS * K_NEI)   // 262144 edge slots

typedef float v2f __attribute__((ext_vector_type(2)));
typedef float v8f __attribute__((ext_vector_type(8)));

__global__ __launch_bounds__(64)
void seg_kernel(const int* __restrict__ batch, int n, int* __restrict__ seg) {
    int m = threadIdx.x;              // molecule id 0..63
    int lo = 0, hi = n;               // lower_bound(batch, m)
    while (lo < hi) { int mid = (lo + hi) >> 1; if (batch[mid] < m) lo = mid + 1; else hi = mid; }
    seg[m] = lo;
    lo = 0; hi = n;                   // lower_bound(batch, m+1)
    while (lo < hi) { int mid = (lo + hi) >> 1; if (batch[mid] < m + 1) lo = mid + 1; else hi = mid; }
    seg[N_MOLS + m] = lo;
}

__global__ __launch_bounds__(256)
void radius_kernel(const float* __restrict__ pos,
                   const int*   __restrict__ seg,
                   float*       __restrict__ out) {
    __shared__ float xs[MPAD], ys[MPAD], zs[MPAD], sq[MPAD];
    __shared__ float strip[16 * MPAD];   // 16 rows x MPAD cols of d^2

    const int mol   = blockIdx.x;
    const int start = seg[mol];
    const int end   = seg[N_MOLS + mol];
    int M = end - start;
    if (M > MPAD) M = MPAD;

    const int tid  = threadIdx.x;
    const int lane = tid & 31;
    // wave index is uniform across the wave: pin it to an SGPR so all loops
    // enclosing the WMMA use scalar control flow (EXEC untouched).
    const int wave = __builtin_amdgcn_readfirstlane(tid >> 5);   // 0..7
    const int half = lane >> 4;         // 0: K=0,1 lanes; 1: K=2,3 lanes
    const int l16  = lane & 15;

    // ---- stage molecule positions into LDS (pad with far-away sentinels) ----
    for (int t = tid; t < MPAD; t += 256) {
        float x, y, z;
        if (t < M) {
            const float* p = pos + (size_t)(start + t) * 3;
            x = p[0]; y = p[1]; z = p[2];
        } else {
            x = 1.0e6f + (float)t; y = x; z = x;   // guaranteed outside cutoff
        }
        xs[t] = x; ys[t] = y; zs[t] = z;
        sq[t] = x * x + y * y + z * z;
    }
    __syncthreads();

    int*   outi   = reinterpret_cast<int*>(out);
    const int nrt = (M + 15) >> 4;      // row tiles == col tiles
    const int colLimit = nrt << 4;

    for (int it = 0; it < nrt; ++it) {
        const int i0 = it << 4;

        // A-matrix regs (16x4 f32): row m = [-2x, -2y, -2z, 1]
        // lanes 0-15 hold K=0,1 ; lanes 16-31 hold K=2,3  (ISA layout)
        v2f a;
        a.x = half ? (-2.0f * zs[i0 + l16]) : (-2.0f * xs[i0 + l16]);
        a.y = half ? 1.0f                    : (-2.0f * ys[i0 + l16]);

        // ---- compute d^2 strip: rows [i0,i0+16) x cols [0,colLimit) ----
        // jt/nrt are SGPR values -> scalar loop, EXEC all-1s at the WMMA.
        for (int jt = wave; jt < nrt; jt += 8) {
            const int c0 = jt << 4;
            // B-matrix regs (4x16 f32): col n = [x; y; z; |p|^2]
            v2f b;
            b.x = half ? zs[c0 + l16] : xs[c0 + l16];
            b.y = half ? sq[c0 + l16] : ys[c0 + l16];

            v8f d = {0.f, 0.f, 0.f, 0.f, 0.f, 0.f, 0.f, 0.f};
            // D(m,n) = -2 p_i(m) . p_j(n) + |p_j(n)|^2
            d = __builtin_amdgcn_wmma_f32_16x16x4_f32(
                    false, a, false, b, (short)0, d, false, false);

            // add |p_i|^2 and spill tile into the LDS strip
            #pragma unroll
            for (int r = 0; r < 8; ++r) {
                const int row = r + (half << 3);      // C/D layout: VGPR r -> row r / r+8
                strip[row * MPAD + (c0 + l16)] = d[r] + sq[i0 + row];
            }
        }
        __syncthreads();

        // ---- top-K selection: wave w owns rows 2w, 2w+1 of this tile ----
        for (int rr = 0; rr < 2; ++rr) {
            const int rowLocal = (wave << 1) | rr;    // SGPR-derived
            const int iLocal   = i0 + rowLocal;
            if (iLocal >= M) break;                   // scalar branch
            const int i = start + iLocal;
            const float* rowp = &strip[rowLocal * MPAD];

            unsigned taken = 0u;                      // per-lane strided-col bitmask
            for (int k = 0; k < K_NEI; ++k) {
                float best = 3.0e38f;
                int   bidx = -1;
                for (int c = lane, t = 0; c < colLimit; c += 32, ++t) {
                    const float v = rowp[c];
                    const bool ok = (c < M) && (c != iLocal) && (v <= CUTOFF2)
                                    && !((taken >> t) & 1u);
                    if (ok && v < best) { best = v; bidx = c; }
                }
                // wave32 min-reduce on (best, bidx), tie -> smaller index
                #pragma unroll
                for (int off = 16; off > 0; off >>= 1) {
                    const float ov = __shfl_xor(best, off, 32);
                    const int   oi = __shfl_xor(bidx, off, 32);
                    if (ov < best || (ov == best && (unsigned)oi < (unsigned)bidx)) {
                        best = ov; bidx = oi;
                    }
                }

                const int e = i * K_NEI + k;
                if (bidx >= 0) {
                    if ((bidx & 31) == lane) taken |= 1u << (bidx >> 5);
                    if (lane == 0) {
                        outi[e]      = i;
                        outi[NK + e] = start + bidx;
                        float* vec = out + 2 * NK + 3 * (size_t)e;
                        vec[0] = xs[bidx] - xs[iLocal];
                        vec[1] = ys[bidx] - ys[iLocal];
                        vec[2] = zs[bidx] - zs[iLocal];
                        out[5 * (size_t)NK + e] = 1.0f;   // mask true
                    }
                } else if (lane == 0) {
                    outi[e]      = i;                     // pad -> self edge
                    outi[NK + e] = i;
                    float* vec = out + 2 * NK + 3 * (size_t)e;
                    vec[0] = 0.f; vec[1] = 0.f; vec[2] = 0.f;
                    out[5 * (size_t)NK + e] = 0.f;        // mask false
                }
            }
        }
        __syncthreads();   // strip reused by next row tile
    }
}

extern "C" void kernel_launch(void* const* d_in, const int* in_sizes, int n_in,
                              void* d_out, int out_size, void* d_ws, size_t ws_size,
                              hipStream_t stream) {
    const float* pos   = (const float*)d_in[0];   // [8192,3] f32
    const int*   batch = (const int*)d_in[1];     // [8192]   i32 (sorted)
    const int    n     = in_sizes[1];

    int* seg = (int*)d_ws;                        // [128] ints: starts, ends
    seg_kernel<<<1, N_MOLS, 0, stream>>>(batch, n, seg);
    radius_kernel<<<N_MOLS, 256, 0, stream>>>(pos, seg, (float*)d_out);
}